// GuidedFilter2d_41532333753082
// MI455X (gfx1250) — compile-verified
//
#include <hip/hip_runtime.h>
#include <hip/hip_bf16.h>
#include <stdint.h>

// Guided filter, radius 30, eps 1e-4, on 12 planes of 1024x1024 f32.
// Memory-bound problem: sliding-window box sums (O(1) adds/pixel) + analytic N.
// CDNA5 path: async global->LDS staging (ASYNCcnt) for the haloed tiles.

#define W_    1024
#define H_    1024
#define RAD   30
#define EPSF  0.0001f
#define TILE  64
#define REG   124            // TILE + 2*RAD padded region
#define GSTR  130            // LDS row stride (floats) for raw staged tiles
#define HSTR  66             // LDS row stride (floats) for h-sum tiles
#define NPL   12

#ifndef __has_builtin
#define __has_builtin(x) 0
#endif

typedef __attribute__((address_space(1))) int as1_int;   // global
typedef __attribute__((address_space(3))) int as3_int;   // LDS

#if __has_builtin(__builtin_amdgcn_global_load_async_to_lds_b32)
#define ASYNC_COPY_B32(gsrc, ldst) \
  __builtin_amdgcn_global_load_async_to_lds_b32((as1_int*)(gsrc), (as3_int*)(ldst), 0, 0)
#else
#define ASYNC_COPY_B32(gsrc, ldst)                                          \
  do {                                                                      \
    unsigned _lofs = (unsigned)(size_t)(as3_int*)(ldst);                    \
    asm volatile("global_load_async_to_lds_b32 %0, %1, off"                 \
                 :: "v"(_lofs), "v"((const void*)(gsrc)) : "memory");       \
  } while (0)
#endif

#if __has_builtin(__builtin_amdgcn_s_wait_asynccnt)
#define ASYNC_WAIT0() __builtin_amdgcn_s_wait_asynccnt(0)
#else
#define ASYNC_WAIT0() asm volatile("s_wait_asynccnt 0" ::: "memory")
#endif

__device__ __forceinline__ float clamp_count(int g) {
  int lo = g - RAD; if (lo < 0) lo = 0;
  int hi = g + RAD; if (hi > 1023) hi = 1023;
  return (float)(hi - lo + 1);
}

// -------- Stage 1: box sums of {g, x, g*x, g*g} -> a, b -------------------
__global__ __launch_bounds__(256) void gf_stage1(
    const float* __restrict__ guide, const float* __restrict__ x,
    float* __restrict__ A, float* __restrict__ Bv)
{
  __shared__ float gbuf[REG * GSTR];
  __shared__ float xbuf[REG * GSTR];
  __shared__ float hsg [REG * HSTR];
  __shared__ float hsx [REG * HSTR];
  __shared__ float hsgx[REG * HSTR];
  __shared__ float hsgg[REG * HSTR];

  const int tid = threadIdx.x;
  const int tx0 = blockIdx.x * TILE;
  const int ty0 = blockIdx.y * TILE;
  const size_t pbase = (size_t)blockIdx.z * (size_t)(W_ * H_);

  // zero-fill (supplies the clipped-window zero padding)
  for (int i = tid; i < REG * GSTR; i += 256) { gbuf[i] = 0.0f; xbuf[i] = 0.0f; }
  __syncthreads();

  // async-stage the in-image part of the 124x124 halo region (coalesced)
  for (int i = tid; i < REG * REG; i += 256) {
    const int rr = i / REG;
    const int cc = i - rr * REG;
    const int gy = ty0 - RAD + rr;
    const int gc = tx0 - RAD + cc;
    if ((unsigned)gy < (unsigned)H_ && (unsigned)gc < (unsigned)W_) {
      const size_t src = pbase + (size_t)gy * W_ + gc;
      const int dst = rr * GSTR + cc;
      ASYNC_COPY_B32(guide + src, &gbuf[dst]);
      ASYNC_COPY_B32(x     + src, &xbuf[dst]);
    }
  }
  ASYNC_WAIT0();
  __syncthreads();

  // horizontal sliding 61-window; thread = (row, half-row)
  {
    const int rr = tid >> 1;
    if (rr < REG) {
      const int c0 = (tid & 1) * 32;
      const float* gr = &gbuf[rr * GSTR];
      const float* xr = &xbuf[rr * GSTR];
      float sg = 0.f, sx = 0.f, sgx = 0.f, sgg = 0.f;
      for (int k = c0; k < c0 + 61; ++k) {
        const float g = gr[k], xx = xr[k];
        sg += g; sx += xx; sgx += g * xx; sgg += g * g;
      }
      const int hb = rr * HSTR;
      for (int i = 0; i < 32; ++i) {
        const int c = c0 + i;
        hsg[hb + c] = sg; hsx[hb + c] = sx; hsgx[hb + c] = sgx; hsgg[hb + c] = sgg;
        if (i < 31) {
          const float ga = gr[c + 61], xa = xr[c + 61];
          const float gs = gr[c],      xs = xr[c];
          sg  += ga - gs;            sx  += xa - xs;
          sgx += ga * xa - gs * xs;  sgg += ga * ga - gs * gs;
        }
      }
    }
  }
  __syncthreads();

  // vertical sliding 61-window; thread = (col, 16-row segment)
  {
    const int c  = tid & 63;
    const int r0 = (tid >> 6) * 16;
    float Sg = 0.f, Sx = 0.f, Sgx = 0.f, Sgg = 0.f;
    for (int k = r0; k < r0 + 61; ++k) {
      const int o = k * HSTR + c;
      Sg += hsg[o]; Sx += hsx[o]; Sgx += hsgx[o]; Sgg += hsgg[o];
    }
    const int gc = tx0 + c;
    const float Nx = clamp_count(gc);
    for (int i = 0; i < 16; ++i) {
      const int r  = r0 + i;
      const int gy = ty0 + r;
      const float invN = 1.0f / (Nx * clamp_count(gy));
      const float mI  = Sg  * invN;
      const float mp  = Sx  * invN;
      const float mIp = Sgx * invN;
      const float mII = Sgg * invN;
      const float cov = mIp - mI * mp;
      const float var = mII - mI * mI;
      const float a   = cov / (var + EPSF);
      const float b   = mp - a * mI;
      const size_t o  = pbase + (size_t)gy * W_ + gc;
      A[o] = a; Bv[o] = b;
      if (i < 15) {
        const int ko = (r + 61) * HSTR + c;
        const int kn = r * HSTR + c;
        Sg  += hsg[ko]  - hsg[kn];
        Sx  += hsx[ko]  - hsx[kn];
        Sgx += hsgx[ko] - hsgx[kn];
        Sgg += hsgg[ko] - hsgg[kn];
      }
    }
  }
}

// -------- Stage 2: box sums of {a, b} -> q = mean_a*guide + mean_b ---------
__global__ __launch_bounds__(256) void gf_stage2(
    const float* __restrict__ A, const float* __restrict__ Bv,
    const float* __restrict__ guide, float* __restrict__ q)
{
  __shared__ float abuf[REG * GSTR];
  __shared__ float bbuf[REG * GSTR];
  __shared__ float hsa [REG * HSTR];
  __shared__ float hsb [REG * HSTR];

  const int tid = threadIdx.x;
  const int tx0 = blockIdx.x * TILE;
  const int ty0 = blockIdx.y * TILE;
  const size_t pbase = (size_t)blockIdx.z * (size_t)(W_ * H_);

  for (int i = tid; i < REG * GSTR; i += 256) { abuf[i] = 0.0f; bbuf[i] = 0.0f; }
  __syncthreads();

  for (int i = tid; i < REG * REG; i += 256) {
    const int rr = i / REG;
    const int cc = i - rr * REG;
    const int gy = ty0 - RAD + rr;
    const int gc = tx0 - RAD + cc;
    if ((unsigned)gy < (unsigned)H_ && (unsigned)gc < (unsigned)W_) {
      const size_t src = pbase + (size_t)gy * W_ + gc;
      const int dst = rr * GSTR + cc;
      ASYNC_COPY_B32(A  + src, &abuf[dst]);
      ASYNC_COPY_B32(Bv + src, &bbuf[dst]);
    }
  }
  ASYNC_WAIT0();
  __syncthreads();

  {
    const int rr = tid >> 1;
    if (rr < REG) {
      const int c0 = (tid & 1) * 32;
      const float* ar = &abuf[rr * GSTR];
      const float* br = &bbuf[rr * GSTR];
      float sa = 0.f, sb = 0.f;
      for (int k = c0; k < c0 + 61; ++k) { sa += ar[k]; sb += br[k]; }
      const int hb = rr * HSTR;
      for (int i = 0; i < 32; ++i) {
        const int c = c0 + i;
        hsa[hb + c] = sa; hsb[hb + c] = sb;
        if (i < 31) {
          sa += ar[c + 61] - ar[c];
          sb += br[c + 61] - br[c];
        }
      }
    }
  }
  __syncthreads();

  {
    const int c  = tid & 63;
    const int r0 = (tid >> 6) * 16;
    float Sa = 0.f, Sb = 0.f;
    for (int k = r0; k < r0 + 61; ++k) {
      const int o = k * HSTR + c;
      Sa += hsa[o]; Sb += hsb[o];
    }
    const int gc = tx0 + c;
    const float Nx = clamp_count(gc);
    for (int i = 0; i < 16; ++i) {
      const int r  = r0 + i;
      const int gy = ty0 + r;
      const float invN = 1.0f / (Nx * clamp_count(gy));
      const size_t o = pbase + (size_t)gy * W_ + gc;
      q[o] = (Sa * invN) * guide[o] + (Sb * invN);
      if (i < 15) {
        const int ko = (r + 61) * HSTR + c;
        const int kn = r * HSTR + c;
        Sa += hsa[ko] - hsa[kn];
        Sb += hsb[ko] - hsb[kn];
      }
    }
  }
}

extern "C" void kernel_launch(void* const* d_in, const int* in_sizes, int n_in,
                              void* d_out, int out_size, void* d_ws, size_t ws_size,
                              hipStream_t stream) {
  (void)in_sizes; (void)n_in; (void)out_size; (void)ws_size;
  const float* x     = (const float*)d_in[0];  // setup_inputs order: x, guide
  const float* guide = (const float*)d_in[1];
  float* q = (float*)d_out;

  const size_t n = (size_t)NPL * W_ * H_;      // 12M floats per plane-stack
  float* A  = (float*)d_ws;                    // a  : 48 MB
  float* Bv = A + n;                           // b  : 48 MB  (96 MB total ws)

  dim3 grid(W_ / TILE, H_ / TILE, NPL);        // 16 x 16 x 12 tiles
  dim3 block(256);                             // 8 wave32 per WG

  gf_stage1<<<grid, block, 0, stream>>>(guide, x, A, Bv);
  gf_stage2<<<grid, block, 0, stream>>>(A, Bv, guide, q);
}